// LossFunction_41575283425857
// MI455X (gfx1250) — compile-verified
//
#include <hip/hip_runtime.h>
#include <hip/hip_bf16.h>
#include <math.h>

typedef __attribute__((ext_vector_type(2))) float v2f;
typedef __attribute__((ext_vector_type(8))) float v8f;

#define NJ   4096
#define NZk  32
#define NT   256
#define NSEL 65536
#define HID  128
#define NTILES (NSEL / 16) /* 4096 */

__device__ __forceinline__ float fast_tanh(float x) {
#if __has_builtin(__builtin_amdgcn_tanhf)
  return __builtin_amdgcn_tanhf(x);   // v_tanh_f32 on gfx1250
#else
  return tanhf(x);
#endif
}

__global__ __launch_bounds__(256) void fused_scan_mlp(
    const float* __restrict__ xi0, const float* __restrict__ z_path,
    const float* __restrict__ a_path, const float* __restrict__ A_path,
    const float* __restrict__ W1, const float* __restrict__ b1,
    const float* __restrict__ W2, const float* __restrict__ b2,
    const int* __restrict__ index, float* __restrict__ partial)
{
  const int lane = threadIdx.x & 31;
  const int tile = blockIdx.x * (blockDim.x >> 5) + (threadIdx.x >> 5);
  const int ln   = lane & 15;
  const bool lo  = lane < 16;

  // This lane's selection slot (lanes m and m+16 share slot m of the tile).
  const int s = tile * 16 + ln;
  const int j = index[2 * s + 0];
  const int k = index[2 * s + 1];

  // Per-lane stream pointers: lanes 0-15 stream the a_path row (K=0,1 of the
  // WMMA A-matrix = a_sel twice); lanes 16-31 stream z (K=2) and A (K=3).
  const float* p0;
  const float* p1;
  if (lo) {
    const float* ar = a_path + ((size_t)k * NJ + j) * NT;
    p0 = ar; p1 = ar;
  } else {
    p0 = z_path + (size_t)k * NT;
    p1 = A_path + (size_t)k * NT;
  }

  // B operand = W1 (4x128) in 8 chunks of 4x16; C operand = b1 splat.
  v2f   w1b[8];
  float w2v[8];
  v8f   cmat[8];
#pragma unroll
  for (int c = 0; c < 8; ++c) {
    const int col = c * 16 + ln;
    w1b[c].x = lo ? W1[0 * HID + col] : W1[2 * HID + col]; // K=0 / K=2
    w1b[c].y = lo ? W1[1 * HID + col] : W1[3 * HID + col]; // K=1 / K=3
    w2v[c]   = W2[col];
    const float bb = b1[col];
#pragma unroll
    for (int r = 0; r < 8; ++r) cmat[c][r] = bb;
  }
  const float b2v = b2[0];

  float xi = xi0[(size_t)j * NZk + k];  // meaningful in lanes 0..15
  float loss_acc = 0.0f;

  // After the register-halving fold, lane l holds x2 for tile row Msrc.
  const int Mloc = ((lane & 1) << 2) | (lane & 2) | ((lane >> 2) & 1);
  const int Msrc = (lo ? 0 : 8) + Mloc;

  auto step = [&](float v0, float v1) {
    // xi recurrence: lanes<16 have a_val in v0; fetch z,A from partner lane.
    const float zf = __shfl_xor(v0, 16, 32);
    const float Af = __shfl_xor(v1, 16, 32);
    xi = 0.95f * xi + fast_tanh(v0 + 0.1f * zf + 0.05f * Af);

    v2f amat; amat.x = v0; amat.y = v1;

    float acc[8];
#pragma unroll
    for (int r = 0; r < 8; ++r) acc[r] = 0.0f;

#pragma unroll
    for (int c = 0; c < 8; ++c) {
      // h_pre(16x16) = feats(16x4) x W1chunk(4x16) + b1chunk
      v8f d = __builtin_amdgcn_wmma_f32_16x16x4_f32(
          false, amat, false, w1b[c], (short)0, cmat[c], false, false);
#pragma unroll
      for (int r = 0; r < 8; ++r)
        acc[r] = fmaf(fast_tanh(d[r]), w2v[c], acc[r]);
    }

    // Butterfly fold: 8 regs x 16 lanes -> 1 reg/lane (x2 per tile row).
    float f4[4], f2[2], f1;
#pragma unroll
    for (int r = 0; r < 4; ++r) {
      const float t    = (lane & 1) ? acc[r] : acc[r + 4];
      const float recv = __shfl_xor(t, 1, 32);
      const float keep = (lane & 1) ? acc[r + 4] : acc[r];
      f4[r] = keep + recv;
    }
#pragma unroll
    for (int r = 0; r < 2; ++r) {
      const float t    = (lane & 2) ? f4[r] : f4[r + 2];
      const float recv = __shfl_xor(t, 2, 32);
      const float keep = (lane & 2) ? f4[r + 2] : f4[r];
      f2[r] = keep + recv;
    }
    {
      const float t    = (lane & 4) ? f2[0] : f2[1];
      const float recv = __shfl_xor(t, 4, 32);
      const float keep = (lane & 4) ? f2[1] : f2[0];
      f1 = keep + recv;
    }
    f1 += __shfl_xor(f1, 8, 32);

    const float x2   = f1 + b2v;
    const float x1   = __shfl(xi, Msrc, 32);  // xi lives in lanes 0..15
    const float diff = x1 - x2;
    loss_acc = fmaf(diff, diff, loss_acc);    // each row counted by 2 lanes
  };

  // t = 254, 253, 252 (scalar), then t = 251..0 as float4 groups (16B aligned).
  for (int t = 254; t >= 252; --t) step(p0[t], p1[t]);
  for (int tb = 248; tb >= 0; tb -= 4) {
    const float4 va = *(const float4*)(p0 + tb);
    const float4 vb = *(const float4*)(p1 + tb);
    step(va.w, vb.w);
    step(va.z, vb.z);
    step(va.y, vb.y);
    step(va.x, vb.x);
  }

#pragma unroll
  for (int m = 1; m < 32; m <<= 1) loss_acc += __shfl_xor(loss_acc, m, 32);
  if (lane == 0) partial[tile] = loss_acc;
}

__global__ __launch_bounds__(256) void reduce_partials(
    const float* __restrict__ partial, float* __restrict__ out)
{
  __shared__ float sm[256];
  float s = 0.0f;
  for (int i = threadIdx.x; i < NTILES; i += 256) s += partial[i];
  sm[threadIdx.x] = s;
  __syncthreads();
  for (int m = 128; m > 0; m >>= 1) {
    if ((int)threadIdx.x < m) sm[threadIdx.x] += sm[threadIdx.x + m];
    __syncthreads();
  }
  if (threadIdx.x == 0) {
    // mean over NSEL per step, sum over 255 steps, /255; each diff^2 counted 2x.
    out[0] = sm[0] * (float)(0.5 / (65536.0 * 255.0));
  }
}

extern "C" void kernel_launch(void* const* d_in, const int* in_sizes, int n_in,
                              void* d_out, int out_size, void* d_ws, size_t ws_size,
                              hipStream_t stream) {
  (void)in_sizes; (void)n_in; (void)out_size; (void)ws_size;
  const float* xi0    = (const float*)d_in[0];
  const float* z_path = (const float*)d_in[1];
  const float* a_path = (const float*)d_in[2];
  const float* A_path = (const float*)d_in[3];
  const float* W1     = (const float*)d_in[4];
  const float* b1     = (const float*)d_in[5];
  const float* W2     = (const float*)d_in[6];
  const float* b2     = (const float*)d_in[7];
  const int*   index  = (const int*)d_in[8];
  float* partial = (float*)d_ws;      // NTILES floats = 16 KB scratch

  // 4096 tiles of 16 selections, 1 wave per tile, 8 waves per block.
  fused_scan_mlp<<<NTILES / 8, 256, 0, stream>>>(
      xi0, z_path, a_path, A_path, W1, b1, W2, b2, index, partial);
  reduce_partials<<<1, 256, 0, stream>>>(partial, (float*)d_out);
}